// HGCL_32246614458469
// MI455X (gfx1250) — compile-verified
//
#include <hip/hip_runtime.h>
#include <cstddef>

// ---------------------------------------------------------------------------
// HGCL hetero-GATv2 for MI455X (gfx1250, wave32).
// Dense projections run on the matrix pipe via V_WMMA_F32_16X16X4_F32 (full
// f32 precision, matching the f32 reference). Edge softmax/scatter phase is
// wave-per-edge with float atomics; it is the L2-bandwidth-bound part.
// ---------------------------------------------------------------------------

typedef __attribute__((ext_vector_type(2))) float v2f;
typedef __attribute__((ext_vector_type(8))) float v8f;

#define N_NEWS 50000
#define N_ENT  100000
#define N_TOP  500
#define DIM_D  256
#define DIM_H0 128
#define DIM_H1 64
#define DIM_OUT 64
#define LRELU_SLOPE 0.2f

// ---------------------------------------------------------------------------
// GEMM: Y[M,N] = X[M,K] @ W[K,N] + bias[N]
// One wave (32 lanes) computes a 16x64 strip: 4 WMMA 16x16 accumulators share
// one A fragment per K-step of 4.  K in {256,128,64}, N in {128,64}.
// A frag (32-bit 16x4): lanes 0-15 rows M, half selects K pair (ISA 7.12.2).
// B frag mirrored: lane -> N column, VGPR+half -> K.
// C/D: VGPR j, lane L -> element (m0 + j + 8*(L>>4), n0 + (L&15)).
// ---------------------------------------------------------------------------
__global__ void gemm_bias_wmma(const float* __restrict__ X,
                               const float* __restrict__ W,
                               const float* __restrict__ bias,
                               float* __restrict__ Y,
                               int M, int K, int N) {
  const int lane = threadIdx.x;        // 0..31, full wave always active (WMMA needs EXEC all-1s)
  const int l15  = lane & 15;
  const int half = lane >> 4;
  const int m0 = blockIdx.x << 4;
  const int n0 = blockIdx.y << 6;
  int arow = m0 + l15;
  if (arow >= M) arow = M - 1;         // clamp loads; stores are guarded below
  const int koff = half << 1;

  v8f acc0 = {}, acc1 = {}, acc2 = {}, acc3 = {};
  const float* xrow = X + (size_t)arow * K;
  for (int k = 0; k < K; k += 4) {
    v2f a;
    { const float2 t = *(const float2*)(xrow + k + koff); a.x = t.x; a.y = t.y; }
    const float* w0 = W + (size_t)(k + koff) * N + n0 + l15;
    const float* w1 = w0 + N;
    v2f b0, b1, b2, b3;
    b0.x = w0[0];  b0.y = w1[0];
    b1.x = w0[16]; b1.y = w1[16];
    b2.x = w0[32]; b2.y = w1[32];
    b3.x = w0[48]; b3.y = w1[48];
    acc0 = __builtin_amdgcn_wmma_f32_16x16x4_f32(false, a, false, b0, (short)0, acc0, false, false);
    acc1 = __builtin_amdgcn_wmma_f32_16x16x4_f32(false, a, false, b1, (short)0, acc1, false, false);
    acc2 = __builtin_amdgcn_wmma_f32_16x16x4_f32(false, a, false, b2, (short)0, acc2, false, false);
    acc3 = __builtin_amdgcn_wmma_f32_16x16x4_f32(false, a, false, b3, (short)0, acc3, false, false);
  }

  const float bi0 = bias[n0 + l15];
  const float bi1 = bias[n0 + 16 + l15];
  const float bi2 = bias[n0 + 32 + l15];
  const float bi3 = bias[n0 + 48 + l15];
#pragma unroll
  for (int j = 0; j < 8; ++j) {
    const int row = m0 + j + (half << 3);
    if (row < M) {
      float* yr = Y + (size_t)row * N + n0;
      yr[l15]      = acc0[j] + bi0;
      yr[16 + l15] = acc1[j] + bi1;
      yr[32 + l15] = acc2[j] + bi2;
      yr[48 + l15] = acc3[j] + bi3;
    }
  }
}

// ordered-int float atomic max (buffer initialized to -inf)
__device__ __forceinline__ void atomicMaxF32(float* addr, float val) {
  if (val >= 0.0f) atomicMax((int*)addr, __float_as_int(val));
  else             atomicMin((unsigned int*)addr, __float_as_uint(val));
}

// ---------------------------------------------------------------------------
// Pass 1: wave per edge. e = leaky_relu(xl[src]+xr[dst], 0.2) . att
//         wave32 shuffle reduction, then emax[dst] = max(emax[dst], e).
// ---------------------------------------------------------------------------
__global__ void edge_logits(const int* __restrict__ src, const int* __restrict__ dst,
                            const float* __restrict__ xl, const float* __restrict__ xr,
                            const float* __restrict__ att,
                            float* __restrict__ e, float* __restrict__ emax,
                            int E, int H) {
  const int w = (blockIdx.x * blockDim.x + threadIdx.x) >> 5;
  const int lane = threadIdx.x & 31;
  if (w >= E) return;                       // wave-uniform exit
  const int s = src[w], d = dst[w];
  const float* pl = xl + (size_t)s * H;
  const float* pr = xr + (size_t)d * H;
  const int fpl = H >> 5;                   // features per lane (coalesced)
  float sum = 0.0f;
  for (int j = 0; j < fpl; ++j) {
    const int i = lane + (j << 5);
    float h = pl[i] + pr[i];
    h = (h > 0.0f) ? h : LRELU_SLOPE * h;
    sum += h * att[i];
  }
  for (int off = 16; off > 0; off >>= 1) sum += __shfl_xor(sum, off, 32);
  if (lane == 0) { e[w] = sum; atomicMaxF32(emax + d, sum); }
}

// ---------------------------------------------------------------------------
// Pass 2: wave per edge. p = exp(e - emax[dst]); denom[dst] += p;
//         num[dst,:] += p * xl[src,:]
// ---------------------------------------------------------------------------
__global__ void edge_scatter(const int* __restrict__ src, const int* __restrict__ dst,
                             const float* __restrict__ xl,
                             const float* __restrict__ e, const float* __restrict__ emax,
                             float* __restrict__ denom, float* __restrict__ num,
                             int E, int H) {
  const int w = (blockIdx.x * blockDim.x + threadIdx.x) >> 5;
  const int lane = threadIdx.x & 31;
  if (w >= E) return;
  const int s = src[w], d = dst[w];
  const float p = expf(e[w] - emax[d]);
  if (lane == 0) atomicAdd(denom + d, p);
  const float* pl = xl + (size_t)s * H;
  float* pn = num + (size_t)d * H;
  const int fpl = H >> 5;
  for (int j = 0; j < fpl; ++j) {
    const int i = lane + (j << 5);
    atomicAdd(pn + i, p * pl[i]);
  }
}

// acc[n,h] += num[n,h] / max(denom[n], 1e-16) + bias[h]
__global__ void finalize_rel(const float* __restrict__ num, const float* __restrict__ denom,
                             const float* __restrict__ bias, float* __restrict__ acc,
                             int Nd, int H) {
  const size_t idx = (size_t)blockIdx.x * blockDim.x + threadIdx.x;
  if (idx >= (size_t)Nd * H) return;
  const int n = (int)(idx / H);
  const int h = (int)(idx - (size_t)n * H);
  float dn = denom[n];
  dn = (dn > 1e-16f) ? dn : 1e-16f;
  acc[idx] += num[idx] / dn + bias[h];
}

__global__ void fill_f32(float* __restrict__ p, float v, size_t n) {
  const size_t idx = (size_t)blockIdx.x * blockDim.x + threadIdx.x;
  if (idx < n) p[idx] = v;
}

__global__ void relu_inplace(float* __restrict__ p, size_t n) {
  const size_t idx = (size_t)blockIdx.x * blockDim.x + threadIdx.x;
  if (idx < n) { const float v = p[idx]; p[idx] = (v > 0.0f) ? v : 0.0f; }
}

// ---------------------------------------------------------------------------
// Driver. Input order (setup_inputs): 0 x_news, 1 x_entities, 2 x_topic,
// 3 ei_ne, 4 ei_en, 5 ei_nt, 6 ei_tn, 7 ei_ee,
// 8..13 W0l,b0l,W0r,b0r,a0,c0, 14..19 W1l,b1l,W1r,b1r,a1,c1, 20 Wlin, 21 blin.
// ei_* are stacked [2,E]: src at offset 0, dst at offset E.
// Workspace use: ~274 MB, all re-initialized every call (graph-replay safe).
// ---------------------------------------------------------------------------
extern "C" void kernel_launch(void* const* d_in, const int* in_sizes, int n_in,
                              void* d_out, int out_size, void* d_ws, size_t ws_size,
                              hipStream_t stream) {
  (void)in_sizes; (void)n_in; (void)out_size; (void)ws_size;
  float* ws = (float*)d_ws;
  size_t o = 0;
  float* accA[3]; float* accB[3];
  accA[0] = ws + o; o += (size_t)N_NEWS * DIM_H0;
  accA[1] = ws + o; o += (size_t)N_ENT  * DIM_H0;
  accA[2] = ws + o; o += (size_t)N_TOP  * DIM_H0;
  accB[0] = ws + o; o += (size_t)N_NEWS * DIM_H1;
  accB[1] = ws + o; o += (size_t)N_ENT  * DIM_H1;
  accB[2] = ws + o; o += (size_t)N_TOP  * DIM_H1;
  float* xlb   = ws + o; o += (size_t)N_ENT * DIM_H0;   // max rows x max H
  float* xrb   = ws + o; o += (size_t)N_ENT * DIM_H0;
  float* numb  = ws + o; o += (size_t)N_ENT * DIM_H0;
  float* denb  = ws + o; o += N_ENT;
  float* emaxb = ws + o; o += N_ENT;
  float* eb    = ws + o; o += 1000000;                  // max E

  const int nNodes[3] = {N_NEWS, N_ENT, N_TOP};
  // relation order: 0 news->ent, 1 ent->news, 2 news->topic, 3 topic->news, 4 ent->ent
  const int relSrc[5] = {0, 1, 0, 2, 1};
  const int relDst[5] = {1, 0, 2, 0, 1};
  const int relEi[5]  = {3, 4, 5, 6, 7};
  const int relE[5]   = {500000, 500000, 50000, 50000, 1000000};

  for (int l = 0; l < 2; ++l) {
    const int din = (l == 0) ? DIM_D  : DIM_H0;
    const int H   = (l == 0) ? DIM_H0 : DIM_H1;
    const int pb = 8 + l * 6;
    const float* Wl = (const float*)d_in[pb + 0];
    const float* bl = (const float*)d_in[pb + 1];
    const float* Wr = (const float*)d_in[pb + 2];
    const float* br = (const float*)d_in[pb + 3];
    const float* av = (const float*)d_in[pb + 4];
    const float* cv = (const float*)d_in[pb + 5];
    const float* X[3];
    float** ACC;
    if (l == 0) {
      X[0] = (const float*)d_in[0]; X[1] = (const float*)d_in[1]; X[2] = (const float*)d_in[2];
      ACC = accA;
    } else {
      X[0] = accA[0]; X[1] = accA[1]; X[2] = accA[2];
      ACC = accB;
    }
    for (int t = 0; t < 3; ++t)
      (void)hipMemsetAsync(ACC[t], 0, (size_t)nNodes[t] * H * sizeof(float), stream);

    for (int r = 0; r < 5; ++r) {
      const int st = relSrc[r], dt = relDst[r];
      const int Ns = nNodes[st], Nd = nNodes[dt], E = relE[r];
      const int* ei = (const int*)d_in[relEi[r]];
      const int* srcIdx = ei;
      const int* dstIdx = ei + E;

      dim3 gl((Ns + 15) / 16, H / 64);
      gemm_bias_wmma<<<gl, 32, 0, stream>>>(X[st], Wl + (size_t)r * din * H, bl + r * H,
                                            xlb, Ns, din, H);
      dim3 gr((Nd + 15) / 16, H / 64);
      gemm_bias_wmma<<<gr, 32, 0, stream>>>(X[dt], Wr + (size_t)r * din * H, br + r * H,
                                            xrb, Nd, din, H);

      (void)hipMemsetAsync(numb, 0, (size_t)Nd * H * sizeof(float), stream);
      (void)hipMemsetAsync(denb, 0, (size_t)Nd * sizeof(float), stream);
      fill_f32<<<(Nd + 255) / 256, 256, 0, stream>>>(emaxb, -__builtin_inff(), (size_t)Nd);

      const int eblocks = (int)(((long long)E * 32 + 255) / 256);
      edge_logits<<<eblocks, 256, 0, stream>>>(srcIdx, dstIdx, xlb, xrb, av + r * H,
                                               eb, emaxb, E, H);
      edge_scatter<<<eblocks, 256, 0, stream>>>(srcIdx, dstIdx, xlb, eb, emaxb,
                                                denb, numb, E, H);
      const int fblocks = (int)(((size_t)Nd * H + 255) / 256);
      finalize_rel<<<fblocks, 256, 0, stream>>>(numb, denb, cv + r * H, ACC[dt], Nd, H);
    }
    for (int t = 0; t < 3; ++t) {
      const size_t cnt = (size_t)nNodes[t] * H;
      relu_inplace<<<(int)((cnt + 255) / 256), 256, 0, stream>>>(ACC[t], cnt);
    }
  }

  // final linear on news nodes: d_out[50000,64] = accB[0] @ Wlin + blin
  dim3 gf((N_NEWS + 15) / 16, DIM_OUT / 64);
  gemm_bias_wmma<<<gf, 32, 0, stream>>>(accB[0], (const float*)d_in[20],
                                        (const float*)d_in[21], (float*)d_out,
                                        N_NEWS, DIM_H1, DIM_OUT);
}